// AllPairsPoseLoss_76879914599136
// MI455X (gfx1250) — compile-verified
//
#include <hip/hip_runtime.h>
#include <hip/hip_bf16.h>

// ---------------------------------------------------------------------------
// AllPairsPoseLoss for MI455X (gfx1250, wave32).
// f32 throughout (matches reference precision). WMMA f32_16x16x4 used as an
// exact f32 wave-reduction engine (A x ones-B => row sums).
// Deterministic two-level reductions; no atomics; no memset needed (all ws
// slots are written before read).
// ---------------------------------------------------------------------------

#define NUM_EP 16
#define COUNT  512
#define CHUNK  16            // i-rows per block
#define NCHUNK (COUNT / CHUNK)   // 32 blocks per episode
#define NTHREADS 256
#define NWAVES (NTHREADS / 32)

typedef float v2f __attribute__((ext_vector_type(2)));
typedef float v8f __attribute__((ext_vector_type(8)));

struct Q  { float x, y, z, w; };
struct V3 { float x, y, z; };

__device__ __forceinline__ V3 qrot(const Q& q, V3 v) {
    // t = 2 * cross(q.xyz, v);  r = v + q.w*t + cross(q.xyz, t)
    float tx = 2.0f * (q.y * v.z - q.z * v.y);
    float ty = 2.0f * (q.z * v.x - q.x * v.z);
    float tz = 2.0f * (q.x * v.y - q.y * v.x);
    V3 r;
    r.x = v.x + q.w * tx + (q.y * tz - q.z * ty);
    r.y = v.y + q.w * ty + (q.z * tx - q.x * tz);
    r.z = v.z + q.w * tz + (q.x * ty - q.y * tx);
    return r;
}

__device__ __forceinline__ Q qmul(const Q& a, const Q& b) {
    Q r;
    r.w = a.w * b.w - a.x * b.x - a.y * b.y - a.z * b.z;
    r.x = a.w * b.x + a.x * b.w + a.y * b.z - a.z * b.y;
    r.y = a.w * b.y - a.x * b.z + a.y * b.w + a.z * b.x;
    r.z = a.w * b.z + a.x * b.y - a.y * b.x + a.z * b.w;
    return r;
}

// Exact f32 wave32 sum via V_WMMA_F32_16X16X4_F32.
// A (16x4): lane m<16 holds {K0,K1}, lane m+16 holds {K2,K3}. We place the
// per-lane accumulator in VGPR0 and zero in VGPR1, B = all ones, C = 0:
//   D[m][n] = x_m + x_{m+16}  (independent of n)
// Lane n<16:  d[r] = rowsum(r)      -> sum_r d[r] = sum of rows 0..7
// Lane n>=16: d[r] = rowsum(r+8)    -> sum_r d[r] = sum of rows 8..15
// Cross-half add via shfl_xor(16) yields the full 32-lane sum in every lane.
__device__ __forceinline__ float wave_sum_wmma(float x) {
    v2f a; a[0] = x;    a[1] = 0.0f;
    v2f b; b[0] = 1.0f; b[1] = 1.0f;
    v8f c = {};
    v8f d = __builtin_amdgcn_wmma_f32_16x16x4_f32(
        /*neg_a=*/false, a, /*neg_b=*/false, b,
        /*c_mod=*/(short)0, c, /*reuse_a=*/false, /*reuse_b=*/false);
    float s = d[0] + d[1] + d[2] + d[3] + d[4] + d[5] + d[6] + d[7];
    s += __shfl_xor(s, 16, 32);
    return s;
}

// Deterministic block(256) sum. Result valid on thread 0.
__device__ __forceinline__ float block_sum(float x, float* sh /*[NWAVES]*/) {
    float ws = wave_sum_wmma(x);
    int lane = threadIdx.x & 31;
    int wave = threadIdx.x >> 5;
    if (lane == 0) sh[wave] = ws;
    __syncthreads();
    float total = 0.0f;
    if (threadIdx.x == 0) {
        for (int w = 0; w < NWAVES; ++w) total += sh[w];
    }
    __syncthreads();   // allow sh reuse by the next call
    return total;
}

// ---------------------------------------------------------------------------
// Pass 1: per-episode reductions dot_pp, dot_pg, g_mag over upper-tri pairs.
// grid = NUM_EP * NCHUNK blocks; block b -> ep = b/NCHUNK, 16 i-rows.
// Thread (ti, tj) = (t%16, t/16): fixed i, strides j by 16.
// ---------------------------------------------------------------------------
__global__ __launch_bounds__(NTHREADS)
void pose_pass1_kernel(const float* __restrict__ pt, const float* __restrict__ pq,
                       const float* __restrict__ gt, const float* __restrict__ gq,
                       float* __restrict__ p1 /*[NUM_EP][NCHUNK][3]*/) {
    const int ep    = blockIdx.x / NCHUNK;
    const int chunk = blockIdx.x % NCHUNK;
    const int ti = threadIdx.x & 15;
    const int tj = threadIdx.x >> 4;
    const int i  = chunk * CHUNK + ti;
    const int gi = ep * COUNT + i;

    V3 pti = { pt[gi*3+0], pt[gi*3+1], pt[gi*3+2] };
    V3 gti = { gt[gi*3+0], gt[gi*3+1], gt[gi*3+2] };
    float4 pqv = ((const float4* __restrict__)pq)[gi];
    float4 gqv = ((const float4* __restrict__)gq)[gi];
    Q pqi = { -pqv.x, -pqv.y, -pqv.z, pqv.w };   // q_inverse
    Q gqi = { -gqv.x, -gqv.y, -gqv.z, gqv.w };

    float app = 0.0f, apg = 0.0f, agm = 0.0f;
    for (int j = tj; j < COUNT; j += 16) {
        if (j <= i) continue;                    // upper triangle only
        const int gj = ep * COUNT + j;
        V3 pd = { pt[gj*3+0] - pti.x, pt[gj*3+1] - pti.y, pt[gj*3+2] - pti.z };
        V3 gd = { gt[gj*3+0] - gti.x, gt[gj*3+1] - gti.y, gt[gj*3+2] - gti.z };
        V3 pr = qrot(pqi, pd);
        V3 gr = qrot(gqi, gd);
        app += pr.x*pr.x + pr.y*pr.y + pr.z*pr.z;
        apg += pr.x*gr.x + pr.y*gr.y + pr.z*gr.z;
        agm += gr.x*gr.x + gr.y*gr.y + gr.z*gr.z;
    }

    __shared__ float sh[NWAVES];
    float spp = block_sum(app, sh);
    float spg = block_sum(apg, sh);
    float sgm = block_sum(agm, sh);
    if (threadIdx.x == 0) {
        float* o = p1 + (ep * NCHUNK + chunk) * 3;
        o[0] = spp; o[1] = spg; o[2] = sgm;
    }
}

// ---------------------------------------------------------------------------
// Per-episode finalize (s_eff, scale_loss) + gravity reduction.
// grid = NUM_EP blocks of 256 threads.
// ---------------------------------------------------------------------------
__global__ __launch_bounds__(NTHREADS)
void pose_finalize_kernel(const float* __restrict__ pq, const float* __restrict__ gq,
                          const float* __restrict__ p1  /*[NUM_EP][NCHUNK][3]*/,
                          float* __restrict__ epp       /*[NUM_EP][2] s_eff,scale*/,
                          float* __restrict__ gravp     /*[NUM_EP] grav sums*/) {
    const int ep = blockIdx.x;

    // gravity: 1 - cos(p_up, g_up), up = (0,1,0)
    float acc = 0.0f;
    for (int c = threadIdx.x; c < COUNT; c += NTHREADS) {
        const int g = ep * COUNT + c;
        float4 pqv = ((const float4* __restrict__)pq)[g];
        float4 gqv = ((const float4* __restrict__)gq)[g];
        Q pquat = { pqv.x, pqv.y, pqv.z, pqv.w };
        Q gquat = { gqv.x, gqv.y, gqv.z, gqv.w };
        V3 up = { 0.0f, 1.0f, 0.0f };
        V3 pu = qrot(pquat, up);
        V3 gu = qrot(gquat, up);
        float dot = pu.x*gu.x + pu.y*gu.y + pu.z*gu.z;
        float np  = sqrtf(pu.x*pu.x + pu.y*pu.y + pu.z*pu.z);
        float ng  = sqrtf(gu.x*gu.x + gu.y*gu.y + gu.z*gu.z);
        float cosv = dot / (fmaxf(np, 1e-8f) * fmaxf(ng, 1e-8f));
        acc += 1.0f - cosv;
    }
    __shared__ float sh[NWAVES];
    float gs = block_sum(acc, sh);

    if (threadIdx.x == 0) {
        gravp[ep] = gs;

        // fixed-order sum of pass-1 partials
        float dpp = 0.0f, dpg = 0.0f, gmag = 0.0f;
        const float* src = p1 + ep * NCHUNK * 3;
        for (int b = 0; b < NCHUNK; ++b) {
            dpp  += src[b*3+0];
            dpg  += src[b*3+1];
            gmag += src[b*3+2];
        }
        bool  small_p = dpp < 1e-5f;
        float s_raw   = dpg / fmaxf(dpp, 1e-12f);
        float s       = small_p ? 1.0f : s_raw;
        float lg      = logf(fabsf(s_raw) + 1e-6f);
        float scale_loss = small_p ? 0.0f : lg * lg;
        s = fminf(fmaxf(s, 0.001f), 1000.0f);
        bool has_scale = gmag > 0.0001f;
        scale_loss = has_scale ? scale_loss : 0.0f;
        float s_eff = has_scale ? s : 1.0f;
        epp[ep*2+0] = s_eff;
        epp[ep*2+1] = scale_loss;
    }
}

__device__ __forceinline__ float smooth_l1(float d) {
    float ad = fabsf(d);
    // BETA = 0.1: 0.5*ad*ad/0.1 = 5*ad*ad  |  ad - 0.05
    return (ad < 0.1f) ? 5.0f * ad * ad : ad - 0.05f;
}

// ---------------------------------------------------------------------------
// Pass 2: trans_loss + rot_loss partials (needs s_eff from finalize).
// Same grid mapping as pass 1.
// ---------------------------------------------------------------------------
__global__ __launch_bounds__(NTHREADS)
void pose_pass2_kernel(const float* __restrict__ pt, const float* __restrict__ pq,
                       const float* __restrict__ gt, const float* __restrict__ gq,
                       const float* __restrict__ epp /*[NUM_EP][2]*/,
                       float* __restrict__ p2        /*[NUM_EP][NCHUNK][2]*/) {
    const int ep    = blockIdx.x / NCHUNK;
    const int chunk = blockIdx.x % NCHUNK;
    const int ti = threadIdx.x & 15;
    const int tj = threadIdx.x >> 4;
    const int i  = chunk * CHUNK + ti;
    const int gi = ep * COUNT + i;
    const float s_eff = epp[ep*2+0];

    V3 pti = { pt[gi*3+0], pt[gi*3+1], pt[gi*3+2] };
    V3 gti = { gt[gi*3+0], gt[gi*3+1], gt[gi*3+2] };
    float4 pqv = ((const float4* __restrict__)pq)[gi];
    float4 gqv = ((const float4* __restrict__)gq)[gi];
    Q pqi = { -pqv.x, -pqv.y, -pqv.z, pqv.w };
    Q gqi = { -gqv.x, -gqv.y, -gqv.z, gqv.w };

    float at = 0.0f, ar = 0.0f;
    for (int j = tj; j < COUNT; j += 16) {
        if (j <= i) continue;
        const int gj = ep * COUNT + j;
        V3 pd = { pt[gj*3+0] - pti.x, pt[gj*3+1] - pti.y, pt[gj*3+2] - pti.z };
        V3 gd = { gt[gj*3+0] - gti.x, gt[gj*3+1] - gti.y, gt[gj*3+2] - gti.z };
        V3 pr = qrot(pqi, pd);
        V3 gr = qrot(gqi, gd);
        at += smooth_l1(pr.x * s_eff - gr.x)
            + smooth_l1(pr.y * s_eff - gr.y)
            + smooth_l1(pr.z * s_eff - gr.z);

        float4 pqj = ((const float4* __restrict__)pq)[gj];
        float4 gqj = ((const float4* __restrict__)gq)[gj];
        Q pb = { pqj.x, pqj.y, pqj.z, pqj.w };
        Q gb = { gqj.x, gqj.y, gqj.z, gqj.w };
        Q prel = qmul(pqi, pb);
        Q grel = qmul(gqi, gb);
        float rd = prel.x*grel.x + prel.y*grel.y + prel.z*grel.z + prel.w*grel.w;
        ar += 1.0f - rd * rd;
    }

    __shared__ float sh[NWAVES];
    float st = block_sum(at, sh);
    float sr = block_sum(ar, sh);
    if (threadIdx.x == 0) {
        float* o = p2 + (ep * NCHUNK + chunk) * 2;
        o[0] = st; o[1] = sr;
    }
}

// ---------------------------------------------------------------------------
// Final combine: 4 output scalars, fixed order.
// ---------------------------------------------------------------------------
__global__ void pose_final_kernel(const float* __restrict__ p2,
                                  const float* __restrict__ epp,
                                  const float* __restrict__ gravp,
                                  float* __restrict__ out) {
    if (threadIdx.x == 0 && blockIdx.x == 0) {
        const float npairs = (float)(COUNT * (COUNT - 1) / 2);   // 130816
        float t = 0.0f, r = 0.0f, g = 0.0f, sc = 0.0f;
        for (int ep = 0; ep < NUM_EP; ++ep) {
            float ts = 0.0f, rs = 0.0f;
            for (int b = 0; b < NCHUNK; ++b) {
                ts += p2[(ep * NCHUNK + b) * 2 + 0];
                rs += p2[(ep * NCHUNK + b) * 2 + 1];
            }
            t  += ts / (npairs * 3.0f);
            r  += rs / npairs;
            g  += gravp[ep] / (float)COUNT;
            sc += epp[ep*2+1];
        }
        out[0] = t  / (float)NUM_EP;
        out[1] = r  / (float)NUM_EP;
        out[2] = g  / (float)NUM_EP;
        out[3] = sc / (float)NUM_EP;
    }
}

extern "C" void kernel_launch(void* const* d_in, const int* in_sizes, int n_in,
                              void* d_out, int out_size, void* d_ws, size_t ws_size,
                              hipStream_t stream) {
    const float* pred_t = (const float*)d_in[0];
    const float* pred_q = (const float*)d_in[1];
    const float* gt_t   = (const float*)d_in[2];
    const float* gt_q   = (const float*)d_in[3];
    // d_in[4] = batch_counts (int32) — constant COUNT per episode; unused.
    float* out = (float*)d_out;

    // Workspace layout (floats), all written before read (no memset needed):
    float* w     = (float*)d_ws;
    float* p1    = w;                              // [NUM_EP][NCHUNK][3]
    float* epp   = p1  + NUM_EP * NCHUNK * 3;      // [NUM_EP][2]
    float* p2    = epp + NUM_EP * 2;               // [NUM_EP][NCHUNK][2]
    float* gravp = p2  + NUM_EP * NCHUNK * 2;      // [NUM_EP]

    dim3 blk(NTHREADS);
    dim3 grid_pairs(NUM_EP * NCHUNK);

    pose_pass1_kernel<<<grid_pairs, blk, 0, stream>>>(pred_t, pred_q, gt_t, gt_q, p1);
    pose_finalize_kernel<<<dim3(NUM_EP), blk, 0, stream>>>(pred_q, gt_q, p1, epp, gravp);
    pose_pass2_kernel<<<grid_pairs, blk, 0, stream>>>(pred_t, pred_q, gt_t, gt_q, epp, p2);
    pose_final_kernel<<<dim3(1), dim3(32), 0, stream>>>(p2, epp, gravp, out);
}